// SimplifiedGINEConv_53077205844582
// MI455X (gfx1250) — compile-verified
//
#include <hip/hip_runtime.h>

#define H 128
#define EPSV 0.0f

typedef __attribute__((ext_vector_type(2))) float v2f;
typedef __attribute__((ext_vector_type(8))) float v8f;

// ---------------------------------------------------------------------------
// Kernel 1: d_out = (1+EPS) * x   (d_out is the scatter accumulator)
// ---------------------------------------------------------------------------
__global__ void gine_init_kernel(const float* __restrict__ x,
                                 float* __restrict__ out, int n4) {
    int i = blockIdx.x * blockDim.x + threadIdx.x;
    if (i >= n4) return;
    float4 v = ((const float4*)x)[i];
    v.x *= (1.0f + EPSV); v.y *= (1.0f + EPSV);
    v.z *= (1.0f + EPSV); v.w *= (1.0f + EPSV);
    ((float4*)out)[i] = v;
}

// ---------------------------------------------------------------------------
// Kernel 2: edge scatter. One 32-lane wave per edge; lane c covers columns
// c, c+32, c+64, c+96. atomic f32 adds land in L2 (accum = 51.2MB < 192MB L2).
// ---------------------------------------------------------------------------
__global__ void gine_scatter_kernel(const float* __restrict__ x,
                                    const long long* __restrict__ ei,
                                    const float* __restrict__ ew,
                                    const float* __restrict__ w_edge,
                                    const float* __restrict__ b_edge,
                                    float* __restrict__ out, int E) {
    int g = blockIdx.x * blockDim.x + threadIdx.x;
    int e = g >> 5;
    int c = g & 31;
    if (e >= E) return;
    long long src = ei[e];
    long long dst = ei[(long long)E + e];
    float w = ew[e];
    const float* xs = x + (size_t)src * H;
    float* od = out + (size_t)dst * H;
#pragma unroll
    for (int j = 0; j < 4; ++j) {
        int col = c + 32 * j;
        float v = xs[col] + w * w_edge[col] + b_edge[col];
        atomicAdd(&od[col], v);
    }
}

// ---------------------------------------------------------------------------
// Kernel 3: fused 2-layer MLP via V_WMMA_F32_16X16X4_F32 (exact fp32).
// One wave owns a 16-row tile, computes all 128 output columns.
//   layer1: h = relu(A @ w1.T + b1)   (h staged through wave-private LDS)
//   layer2: out = h @ w2.T + b2       (written back in place to d_out)
// A-fragment (16x4 f32): lanes 0-15 hold M=lane, K={4kb,4kb+1};
//                        lanes 16-31 hold M=lane-16, K={4kb+2,4kb+3}.
// B-fragment mirrors with N across lanes; B[k][n] = W[n][k] is a contiguous
// float2 in the row-major [out,in] weight matrix.
// ---------------------------------------------------------------------------
__launch_bounds__(128)
__global__ void gine_mlp_kernel(float* __restrict__ io,
                                const float* __restrict__ w1,
                                const float* __restrict__ b1,
                                const float* __restrict__ w2,
                                const float* __restrict__ b2, int ntiles) {
    __shared__ float hbuf[4 * 16 * H];  // 32KB: one 16x128 h tile per wave
    const int wave = threadIdx.x >> 5;
    const int lane = threadIdx.x & 31;
    const int tile = blockIdx.x * 4 + wave;
    if (tile >= ntiles) return;  // uniform per wave; active waves keep EXEC=~0

    const int row0 = tile * 16;
    const int lr   = lane & 15;
    const int half = lane >> 4;          // 0: K lo pair, 1: K hi pair
    float* hw = hbuf + wave * 16 * H;    // wave-private region

    v2f a[32];

    // ---- load A fragments from accumulator rows (d_out) ----
    {
        const float* A = io + (size_t)(row0 + lr) * H + 2 * half;
#pragma unroll
        for (int kb = 0; kb < 32; ++kb)
            a[kb] = *(const v2f*)(A + 4 * kb);
    }

    // ---- layer 1: h = relu(A @ w1.T + b1) -> LDS ----
    for (int nt = 0; nt < 8; ++nt) {
        const int n0 = nt * 16;
        const float bb = b1[n0 + lr];
        v8f c = {bb, bb, bb, bb, bb, bb, bb, bb};
        const float* B = w1 + (size_t)(n0 + lr) * H + 2 * half;
#pragma unroll
        for (int kb = 0; kb < 32; ++kb) {
            v2f b = *(const v2f*)(B + 4 * kb);
            c = __builtin_amdgcn_wmma_f32_16x16x4_f32(
                    false, a[kb], false, b, (short)0, c, false, false);
        }
#pragma unroll
        for (int i = 0; i < 8; ++i) {
            float v = c[i] > 0.0f ? c[i] : 0.0f;          // ReLU
            int m = half ? (i + 8) : i;                   // C layout row
            hw[m * H + n0 + lr] = v;
        }
    }

    // ---- reload A fragments from h (LDS, wave-private; dscnt handled) ----
#pragma unroll
    for (int kb = 0; kb < 32; ++kb)
        a[kb] = *(const v2f*)(hw + lr * H + 4 * kb + 2 * half);

    // ---- layer 2: out = h @ w2.T + b2 -> d_out (in place) ----
    for (int nt = 0; nt < 8; ++nt) {
        const int n0 = nt * 16;
        const float bb = b2[n0 + lr];
        v8f c = {bb, bb, bb, bb, bb, bb, bb, bb};
        const float* B = w2 + (size_t)(n0 + lr) * H + 2 * half;
#pragma unroll
        for (int kb = 0; kb < 32; ++kb) {
            v2f b = *(const v2f*)(B + 4 * kb);
            c = __builtin_amdgcn_wmma_f32_16x16x4_f32(
                    false, a[kb], false, b, (short)0, c, false, false);
        }
#pragma unroll
        for (int i = 0; i < 8; ++i) {
            int m = half ? (i + 8) : i;
            io[(size_t)(row0 + m) * H + n0 + lr] = c[i];
        }
    }
}

// ---------------------------------------------------------------------------
// Launch
// ---------------------------------------------------------------------------
extern "C" void kernel_launch(void* const* d_in, const int* in_sizes, int n_in,
                              void* d_out, int out_size, void* d_ws, size_t ws_size,
                              hipStream_t stream) {
    const float*     x      = (const float*)d_in[0];
    const long long* ei     = (const long long*)d_in[1];  // [2,E] int64: src then dst
    const float*     ew     = (const float*)d_in[2];
    const float*     w_edge = (const float*)d_in[3];
    const float*     b_edge = (const float*)d_in[4];
    const float*     w1     = (const float*)d_in[5];
    const float*     b1     = (const float*)d_in[6];
    const float*     w2     = (const float*)d_in[7];
    const float*     b2     = (const float*)d_in[8];
    float*           out    = (float*)d_out;

    const int N = in_sizes[0] / H;   // 100000
    const int E = in_sizes[2];       // 1600000

    // 1) out = (1+eps)*x
    {
        int n4 = (N * H) / 4;
        int blocks = (n4 + 255) / 256;
        gine_init_kernel<<<blocks, 256, 0, stream>>>(x, out, n4);
    }
    // 2) scatter-add edges (32 lanes per edge)
    {
        long long threads = (long long)E * 32;
        int blocks = (int)((threads + 255) / 256);
        gine_scatter_kernel<<<blocks, 256, 0, stream>>>(x, ei, ew, w_edge, b_edge, out, E);
    }
    // 3) fused MLP (WMMA fp32), in place on d_out
    {
        int ntiles = N / 16;  // 6250, N divisible by 16
        int blocks = (ntiles + 3) / 4;
        gine_mlp_kernel<<<blocks, 128, 0, stream>>>(out, w1, b1, w2, b2, ntiles);
    }
}